// PointNetJointLoss_1374389534997
// MI455X (gfx1250) — compile-verified
//
#include <hip/hip_runtime.h>
#include <hip/hip_bf16.h>

typedef float v2f __attribute__((ext_vector_type(2)));
typedef float v8f __attribute__((ext_vector_type(8)));
typedef int   v2i __attribute__((ext_vector_type(2)));

#define LOG_CLAMP (-100.0f)
#define LOG2E 1.44269504088896340736f
#define LN2   0.69314718055994530942f

// ---------------------------------------------------------------------------
// Kernel 1: seg BCE cost accumulation via V_WMMA_F32_16X16X4_F32.
//
// Two batches are packed per wave: A rows 0-7 = batch (2*pair), rows 8-15 =
// batch (2*pair+1); B cols 0-7 / 8-15 are the corresponding one-hot gt masks.
// C's top-left 8x8 (VGPR r, lanes 0-7) and bottom-right 8x8 (VGPR r, lanes
// 24-31) hold the two batches' t[i][j] = sum_k d[i,k]*onehot(gt[k]==j).
// Per-lane rowsum accumulates lg1m for the -(t + rowsum)/N cost.
//
// Hot-loop math uses bare hardware transcendentals (v_exp_f32/v_log_f32 via
// __builtin_amdgcn_exp2f/_logf, base-2 with fused scale multiplies): the -100
// clamp absorbs log(0) = -inf and inputs are probabilities in [1e-4, 1-1e-4],
// so libm's denorm pre-scaling / inf fixups are dead weight that would push a
// memory-bound kernel (75 MB @ 23.3 TB/s ~3.2 us) into being VALU-bound.
// ---------------------------------------------------------------------------
__global__ __launch_bounds__(256) void seg_cost_wmma(
    const float* __restrict__ logp,  // (16, 8, N) log-probabilities
    const int*   __restrict__ gts,   // (16, N) in [0,8)
    float* __restrict__ ws,          // (8, G, 144) partial records
    int N, int G)
{
    __shared__ float pt[8][2][8][8];  // per-wave t partials
    __shared__ float prs[8][32];      // per-wave per-lane rowsum partials

    const int tid  = threadIdx.x;
    const int w    = tid >> 5;
    const int lane = tid & 31;
    const int m    = lane & 15;   // A: row index 0..15 (== B: column index)
    const int sel  = lane >> 4;   // A/B: K sub-pair selector
    const int q    = m >> 3;      // which batch of the pair
    const int rowA = m & 7;       // slot within batch (A side)
    const int ncol = m & 7;       // one-hot column within batch (B side)

    const int pair = blockIdx.y;          // 0..7  -> batches (2p, 2p+1)
    const int g    = blockIdx.x;          // 0..G-1
    const int wavesPerPair = G * 8;
    const int gw    = g * 8 + w;          // wave index within the pair
    const int elemsPerWave = N / wavesPerPair;  // K elements per row per wave
    const int steps = elemsPerWave >> 2;        // 4 K per WMMA step
    const int kbase = gw * elemsPerWave;

    const float* Arow = logp + ((size_t)((2 * pair + q) * 8 + rowA)) * (size_t)N
                        + kbase + 2 * sel;
    const int*   Gp   = gts  + (size_t)(2 * pair + q) * (size_t)N
                        + kbase + 2 * sel;

    v8f acc = {};
    float rowsum = 0.0f;

    #pragma unroll 4
    for (int s = 0; s < steps; ++s) {
        v2f lp = __builtin_nontemporal_load((const v2f*)(Arow + 4 * s));
        v2i g2 = __builtin_nontemporal_load((const v2i*)(Gp + 4 * s));

        // clamped log(p) and log(1-p); exp of the UNclamped logp (matches ref)
        float lx  = fmaxf(lp.x, LOG_CLAMP);
        float ly  = fmaxf(lp.y, LOG_CLAMP);
        float ex  = __builtin_amdgcn_exp2f(lp.x * LOG2E);        // exp(logp)
        float ey  = __builtin_amdgcn_exp2f(lp.y * LOG2E);
        float l1x = fmaxf(__builtin_amdgcn_logf(1.0f - ex) * LN2, LOG_CLAMP);
        float l1y = fmaxf(__builtin_amdgcn_logf(1.0f - ey) * LN2, LOG_CLAMP);

        v2f a; a.x = lx - l1x; a.y = ly - l1y;     // d = lgp - lg1m
        rowsum += l1x + l1y;

        v2f b; b.x = (g2.x == ncol) ? 1.0f : 0.0f; // one-hot B column
               b.y = (g2.y == ncol) ? 1.0f : 0.0f;

        acc = __builtin_amdgcn_wmma_f32_16x16x4_f32(
                  false, a, false, b, (short)0, acc, false, false);
    }

    // stash meaningful C entries: lanes 0-7 -> batch q=0, lanes 24-31 -> q=1
    if (lane < 8) {
        #pragma unroll
        for (int r = 0; r < 8; ++r) pt[w][0][r][lane] = acc[r];
    } else if (lane >= 24) {
        #pragma unroll
        for (int r = 0; r < 8; ++r) pt[w][1][r][lane - 24] = acc[r];
    }
    prs[w][lane] = rowsum;
    __syncthreads();

    // workgroup reduction -> one 144-float record (fixed order, deterministic)
    float* rec = ws + (size_t)(pair * G + g) * 144;
    if (tid < 128) {
        int qq = tid >> 6, r = (tid >> 3) & 7, j = tid & 7;
        float s = 0.0f;
        #pragma unroll
        for (int ww = 0; ww < 8; ++ww) s += pt[ww][qq][r][j];
        rec[qq * 64 + r * 8 + j] = s;
    } else if (tid < 144) {
        int idx = tid - 128;
        int qq = idx >> 3, r = idx & 7;
        // row (qq, r) lives in lanes qq*8+r (sel=0) and 16+qq*8+r (sel=1)
        float s = 0.0f;
        #pragma unroll
        for (int ww = 0; ww < 8; ++ww)
            s += prs[ww][qq * 8 + r] + prs[ww][16 + qq * 8 + r];
        rec[128 + qq * 8 + r] = s;
    }
}

// ---------------------------------------------------------------------------
// Kernel 2: reduce partials -> seg cost; aff cost directly; 2^8 bitmask DP
// (exact min-over-permutations assignment) for 16 batches x 2 loss types.
// ---------------------------------------------------------------------------
__global__ __launch_bounds__(256) void finalize_kernel(
    const float* __restrict__ ws, int G, int N,
    const float* __restrict__ pred_aff,  // (16, 8, A) probabilities
    const int*   __restrict__ gt_aff,    // (16, 8, A) in {0,1}
    int A,
    float* __restrict__ out)             // [seg, aff, aff + 3*seg]
{
    __shared__ float cs[16][64];
    __shared__ float ca[16][64];
    __shared__ float res[32];
    const int tid = threadIdx.x;

    // seg cost: -(t + rowsum)/N, summing G partial records per pair
    for (int e = tid; e < 16 * 64; e += 256) {
        int b = e >> 6, i = (e >> 3) & 7, j = e & 7;
        int pair = b >> 1, q = b & 1;
        float t = 0.0f, rs = 0.0f;
        const float* base = ws + (size_t)pair * G * 144;
        for (int gg = 0; gg < G; ++gg) {
            const float* rec = base + (size_t)gg * 144;
            t  += rec[q * 64 + i * 8 + j];
            rs += rec[128 + q * 8 + i];
        }
        cs[b][i * 8 + j] = -(t + rs) / (float)N;
    }

    // aff cost (K = A = 18): plain BCE mean (precise libm here; tiny)
    for (int e = tid; e < 16 * 64; e += 256) {
        int b = e >> 6, i = (e >> 3) & 7, j = e & 7;
        const float* pp = pred_aff + ((size_t)b * 8 + i) * A;
        const int*   gg = gt_aff   + ((size_t)b * 8 + j) * A;
        float s = 0.0f;
        for (int k = 0; k < A; ++k) {
            float p  = pp[k];
            float lp = fmaxf(logf(p), LOG_CLAMP);
            float l1 = fmaxf(log1pf(-p), LOG_CLAMP);
            s += gg[k] ? lp : l1;
        }
        ca[b][i * 8 + j] = -s / (float)A;
    }
    __syncthreads();

    // exact assignment: dp over column subsets (== min over 8! permutations)
    if (tid < 32) {
        int b = tid & 15, type = tid >> 4;
        const float* C = (type == 0) ? cs[b] : ca[b];
        float dp[256];
        dp[0] = 0.0f;
        for (int mask = 1; mask < 256; ++mask) {
            int i = __popc(mask) - 1;
            float best = 3.0e38f;
            for (int j = 0; j < 8; ++j) {
                if (mask & (1 << j)) {
                    float v = dp[mask ^ (1 << j)] + C[i * 8 + j];
                    best = fminf(best, v);
                }
            }
            dp[mask] = best;
        }
        res[tid] = dp[255];
    }
    __syncthreads();

    if (tid == 0) {
        float seg = 0.0f, aff = 0.0f;
        for (int b = 0; b < 16; ++b) { seg += res[b]; aff += res[16 + b]; }
        seg *= (1.0f / 16.0f);
        aff *= (1.0f / 16.0f);
        out[0] = seg;
        out[1] = aff;
        out[2] = aff + 3.0f * seg;
    }
}

extern "C" void kernel_launch(void* const* d_in, const int* in_sizes, int n_in,
                              void* d_out, int out_size, void* d_ws, size_t ws_size,
                              hipStream_t stream) {
    const float* pred_seg_logp = (const float*)d_in[0];
    const float* pred_aff      = (const float*)d_in[1];
    const int*   gt_seg_mask   = (const int*)d_in[2];
    const int*   gt_aff        = (const int*)d_in[3];
    float* out = (float*)d_out;
    float* ws  = (float*)d_ws;

    const int B = 16, S = 8;
    const int N = in_sizes[0] / (B * S);   // 131072
    const int A = in_sizes[1] / (B * S);   // 18

    // G blocks per batch-pair; needs (B/2)*G*144 floats of workspace and
    // N divisible by G*8 waves*4 K-per-step.
    int G = 64;
    while (G > 1 && (size_t)(B / 2) * G * 144 * sizeof(float) > ws_size) G >>= 1;
    while (G > 1 && (N % (G * 32)) != 0) G >>= 1;

    dim3 grid(G, B / 2);
    seg_cost_wmma<<<grid, 256, 0, stream>>>(pred_seg_logp, gt_seg_mask, ws, N, G);
    finalize_kernel<<<1, 256, 0, stream>>>(ws, G, N, pred_aff, gt_aff, A, out);
}